// MoPeAttention_65386582114334
// MI455X (gfx1250) — compile-verified
//
#include <hip/hip_runtime.h>
#include <math.h>

#define B_ 4
#define T_ 2048
#define D_ 1024
#define M_ 64
#define TP1 2049
#define CHUNK 32
#define NC (T_ / CHUNK)        // 64 chunks
#define JITTER_ 1e-5f
#define LN_EPS_ 1e-5f

// ---- d_out layout (floats), reference return order ----
#define Y_OFF    ((size_t)0)                    // B*T*D = 8388608
#define CKK_OFF  ((size_t)8388608)              // B*M*M = 16384
#define CVK_OFF  ((size_t)8404992)
#define CVV_OFF  ((size_t)8421376)
#define LOSS_OFF ((size_t)8437760)              // 1
#define Q_OFF    ((size_t)8437761)              // B*T*M = 524288
#define K_OFF    ((size_t)8962049)
#define V_OFF    ((size_t)9486337)

// ---- workspace layout (floats) ----
#define NORM_OFF  ((size_t)0)                                    // B*TP1*D
#define OUTX_OFF  ((size_t)((size_t)B_ * TP1 * D_))              // B*T*M
#define SUMS_OFF  (OUTX_OFF + (size_t)B_ * T_ * M_)              // B*NC*3*M*M
#define LPART_OFF (SUMS_OFF + (size_t)B_ * NC * 3 * M_ * M_)     // B*NC

typedef float v2f __attribute__((ext_vector_type(2)));
typedef float v8f __attribute__((ext_vector_type(8)));

// Native fp32 WMMA on CDNA5: D(16x16) = A(16x4) * B(4x16) + C
__device__ __forceinline__ v8f wmma4(v2f a, v2f b, v8f c) {
  return __builtin_amdgcn_wmma_f32_16x16x4_f32(false, a, false, b, (short)0, c,
                                               false, false);
}

__device__ __forceinline__ float sigmoidf_(float z) {
  return 1.f / (1.f + __expf(-z));
}

// ============================================================
// Kernel 1: LayerNorm of zero-padded sequence -> norm (B,TP1,D)
// ============================================================
__global__ __launch_bounds__(256) void ln_kernel(const float* __restrict__ x,
                                                 const float* __restrict__ gamma,
                                                 const float* __restrict__ beta,
                                                 float* __restrict__ norm) {
  const int row = blockIdx.x;             // 0 .. B*TP1-1
  const int b = row / TP1, t = row % TP1;
  const int tid = threadIdx.x;
  __shared__ float red[256];

  const float* xr = x + ((size_t)b * T_ + (size_t)(t - 1)) * D_;
  float vals[4];
#pragma unroll
  for (int i = 0; i < 4; ++i) {
    int d = tid + i * 256;
    vals[i] = (t == 0) ? 0.f : xr[d];
  }
  float s = vals[0] + vals[1] + vals[2] + vals[3];
  red[tid] = s;
  __syncthreads();
  for (int off = 128; off > 0; off >>= 1) {
    if (tid < off) red[tid] += red[tid + off];
    __syncthreads();
  }
  const float mu = red[0] * (1.f / D_);
  __syncthreads();
  float s2 = 0.f;
#pragma unroll
  for (int i = 0; i < 4; ++i) {
    float dvi = vals[i] - mu;
    s2 += dvi * dvi;
  }
  red[tid] = s2;
  __syncthreads();
  for (int off = 128; off > 0; off >>= 1) {
    if (tid < off) red[tid] += red[tid + off];
    __syncthreads();
  }
  const float rstd = rsqrtf(red[0] * (1.f / D_) + LN_EPS_);
  float* nr = norm + ((size_t)b * TP1 + t) * D_;
#pragma unroll
  for (int i = 0; i < 4; ++i) {
    int d = tid + i * 256;
    nr[d] = (vals[i] - mu) * rstd * gamma[d] + beta[d];
  }
}

// ============================================================
// Kernel 2: lerp-gated q/k/v projection GEMM via fp32 WMMA
// grid (T/16, B, 3), block 128 (4 waves, one 16x16 n-tile each)
// ============================================================
__global__ __launch_bounds__(128) void qkv_kernel(const float* __restrict__ norm,
                                                  const float* __restrict__ lerp_w,
                                                  const float* __restrict__ w,
                                                  float* __restrict__ out) {
  const int t0 = blockIdx.x * 16;
  const int b = blockIdx.y;
  const int i = blockIdx.z;               // 0=q 1=k 2=v
  const int tid = threadIdx.x;
  const int wv = tid >> 5, lane = tid & 31;
  const int row = lane & 15, hi = lane >> 4;

  __shared__ float slin[16][68];
  __shared__ float swt[64][68];
  __shared__ float slw[D_];

  for (int e = tid; e < D_; e += 128) slw[e] = sigmoidf_(lerp_w[(size_t)i * D_ + e]);

  const float* nb = norm + ((size_t)b * TP1) * D_;
  const float* wb = w + (size_t)i * M_ * D_;
  v8f acc = {};

  for (int d0 = 0; d0 < D_; d0 += 64) {
    __syncthreads();
    // stage lin tile (16 rows of t, 64 d): lin = last + (cur-last)*sig
    for (int e = tid; e < 16 * 64; e += 128) {
      int r = e >> 6, dl = e & 63;
      float lastv = nb[(size_t)(t0 + r) * D_ + d0 + dl];
      float curv = nb[(size_t)(t0 + r + 1) * D_ + d0 + dl];
      slin[r][dl] = lastv + (curv - lastv) * slw[d0 + dl];
    }
    // stage weight tile transposed: swt[d_local][m]
    for (int e = tid; e < 64 * 64; e += 128) {
      int m = e >> 6, dl = e & 63;
      swt[dl][m] = wb[(size_t)m * D_ + d0 + dl];
    }
    if (d0 + 64 < D_)
      __builtin_prefetch(&nb[(size_t)(t0 + (tid & 15)) * D_ + d0 + 64], 0, 1);
    __syncthreads();
#pragma unroll
    for (int kq = 0; kq < 16; ++kq) {
      const int k0 = kq * 4 + 2 * hi;
      v2f a;
      a.x = slin[row][k0];
      a.y = slin[row][k0 + 1];
      v2f bb;
      bb.x = swt[k0][wv * 16 + row];
      bb.y = swt[k0 + 1][wv * 16 + row];
      acc = wmma4(a, bb, acc);
    }
  }
  const size_t offs = (i == 0) ? Q_OFF : (i == 1) ? K_OFF : V_OFF;
  float* dst = out + offs + (size_t)b * T_ * M_;
#pragma unroll
  for (int r = 0; r < 8; ++r) {
    int tr = t0 + r + 8 * hi;
    dst[(size_t)tr * M_ + wv * 16 + row] = acc[r];
  }
}

// ============================================================
// Kernel 3: per-chunk outer-product sums  S = sum_t X_t Y_t^T
// iii: 0 -> kk, 1 -> vk, 2 -> vv.  grid (NC, B, 3), block 128.
// ============================================================
__global__ __launch_bounds__(128) void csum_kernel(const float* __restrict__ out,
                                                   float* __restrict__ sums) {
  const int c = blockIdx.x, b = blockIdx.y, iii = blockIdx.z;
  const int tid = threadIdx.x;
  const int wv = tid >> 5, lane = tid & 31;
  const int row = lane & 15, hi = lane >> 4;

  const size_t base = (size_t)(b * T_ + c * CHUNK) * M_;
  const float* X = out + ((iii == 0) ? K_OFF : V_OFF) + base;
  const float* Y = out + ((iii == 2) ? V_OFF : K_OFF) + base;

  __shared__ float sx[CHUNK][68];
  __shared__ float sy[CHUNK][68];
  for (int e = tid; e < CHUNK * 64; e += 128) {
    int tt = e >> 6, m = e & 63;
    sx[tt][m] = X[(size_t)tt * M_ + m];
    sy[tt][m] = Y[(size_t)tt * M_ + m];
  }
  __syncthreads();

  float* S = sums + (((size_t)b * NC + c) * 3 + iii) * (M_ * M_);
  const int m0 = wv * 16;
#pragma unroll
  for (int nt = 0; nt < 4; ++nt) {
    v8f acc = {};
#pragma unroll
    for (int kq = 0; kq < CHUNK / 4; ++kq) {
      const int k0 = kq * 4 + 2 * hi;
      v2f a;                           // A[m][t] = X[t][m]
      a.x = sx[k0][m0 + row];
      a.y = sx[k0 + 1][m0 + row];
      v2f bb;                          // B[t][n] = Y[t][n]
      bb.x = sy[k0][nt * 16 + row];
      bb.y = sy[k0 + 1][nt * 16 + row];
      acc = wmma4(a, bb, acc);
    }
#pragma unroll
    for (int r = 0; r < 8; ++r)
      S[(size_t)(m0 + r + 8 * hi) * M_ + nt * 16 + row] = acc[r];
  }
}

// ============================================================
// Kernel 4: exclusive scan of chunk sums (in place) + write the
// grand totals = C_*[:, -1] directly into d_out. grid (3, B).
// ============================================================
__global__ __launch_bounds__(256) void scan_kernel(float* __restrict__ sums,
                                                   float* __restrict__ out) {
  const int iii = blockIdx.x, b = blockIdx.y;
  const int tid = threadIdx.x;
  float run[16];
#pragma unroll
  for (int j = 0; j < 16; ++j) run[j] = 0.f;
  for (int c = 0; c < NC; ++c) {
    float* S = sums + (((size_t)b * NC + c) * 3 + iii) * (M_ * M_);
#pragma unroll
    for (int j = 0; j < 16; ++j) {
      int e = tid + j * 256;
      float tv = S[e];
      S[e] = run[j];
      run[j] += tv;
    }
  }
  const size_t foff = (iii == 0) ? CKK_OFF : (iii == 1) ? CVK_OFF : CVV_OFF;
  float* F = out + foff + (size_t)b * M_ * M_;
#pragma unroll
  for (int j = 0; j < 16; ++j) F[tid + j * 256] = run[j];
}

// ------------------------------------------------------------
// 64x64x64 fp32 matmul in LDS via WMMA: Dst = A * B
// 8 waves x 2 tiles x 16 k-steps of 16x16x4
// ------------------------------------------------------------
__device__ __forceinline__ void mm64(const float (*A)[65], const float (*Bm)[65],
                                     float (*Dst)[65], int tid) {
  const int wv = tid >> 5, lane = tid & 31;
  const int row = lane & 15, hi = lane >> 4;
  for (int tI = wv; tI < 16; tI += 8) {
    const int ti = tI >> 2, tj = tI & 3;
    v8f acc = {};
#pragma unroll
    for (int kq = 0; kq < 16; ++kq) {
      const int k0 = kq * 4 + 2 * hi;
      v2f a;
      a.x = A[ti * 16 + row][k0];
      a.y = A[ti * 16 + row][k0 + 1];
      v2f bb;
      bb.x = Bm[k0][tj * 16 + row];
      bb.y = Bm[k0 + 1][tj * 16 + row];
      acc = wmma4(a, bb, acc);
    }
#pragma unroll
    for (int r = 0; r < 8; ++r)
      Dst[ti * 16 + r + 8 * hi][tj * 16 + row] = acc[r];
  }
}

// ============================================================
// Kernel 5: per-chunk sequential recurrence.
//  - Gauss-Jordan inverse of (prefix_kk + eps I)  [once]
//  - per t: rank-1 C updates, Sherman-Morrison inverse update,
//    memory = C_vk*inv (WMMA), tmp = memory*C_kk (WMMA),
//    loss scalars, out_x = memory*q.
// grid (NC, B), block 256 (8 waves).
// ============================================================
__global__ __launch_bounds__(256) void pert_kernel(const float* __restrict__ out,
                                                   const float* __restrict__ sums,
                                                   float* __restrict__ outx,
                                                   float* __restrict__ lpart) {
  const int c = blockIdx.x, b = blockIdx.y;
  const int tid = threadIdx.x;
  const int t0 = c * CHUNK;

  __shared__ float Ckk[64][65];
  __shared__ float Cvk[64][65];
  __shared__ float Inv[64][65];
  __shared__ float Mem[64][65];
  __shared__ float Tmp[64][65];   // doubles as GJ scratch
  __shared__ float kv[64], vv_[64], qv[64], uu[64], ff[64];
  __shared__ float red[256];
  __shared__ float sc[2];

  const float* SK = sums + (((size_t)b * NC + c) * 3 + 0) * (M_ * M_);
  const float* SV = sums + (((size_t)b * NC + c) * 3 + 1) * (M_ * M_);
  const float* SW = sums + (((size_t)b * NC + c) * 3 + 2) * (M_ * M_);

  for (int e = tid; e < 64 * 64; e += 256) {
    int r = e >> 6, cc = e & 63;
    float pk = SK[e];
    Ckk[r][cc] = pk;
    Tmp[r][cc] = pk + ((r == cc) ? JITTER_ : 0.f);
    Inv[r][cc] = (r == cc) ? 1.f : 0.f;
    Cvk[r][cc] = SV[e];
  }
  red[tid] = (tid < 64) ? SW[(size_t)tid * 64 + tid] : 0.f;  // trace prefix_vv
  __syncthreads();
  for (int off = 128; off > 0; off >>= 1) {
    if (tid < off) red[tid] += red[tid + off];
    __syncthreads();
  }
  if (tid == 0) sc[0] = red[0];
  __syncthreads();

  // ---- Gauss-Jordan: Inv = inv(Tmp), no pivoting (SPD + jitter) ----
  for (int p = 0; p < 64; ++p) {
    const float pivinv = 1.f / Tmp[p][p];
    __syncthreads();
    if (tid < 64) Tmp[p][tid] *= pivinv;
    else if (tid < 128) Inv[p][tid - 64] *= pivinv;
    if (tid < 64) ff[tid] = (tid == p) ? 0.f : Tmp[tid][p];
    __syncthreads();
    for (int e = tid; e < 64 * 128; e += 256) {
      int r = e >> 7, col = e & 127;
      float f = ff[r];
      if (col < 64) Tmp[r][col] -= f * Tmp[p][col];
      else Inv[r][col - 64] -= f * Inv[p][col - 64];
    }
    __syncthreads();
  }

  const float* qb = out + Q_OFF + (size_t)b * T_ * M_;
  const float* kb = out + K_OFF + (size_t)b * T_ * M_;
  const float* vb = out + V_OFF + (size_t)b * T_ * M_;
  float lsum = 0.f;

  for (int tt = 0; tt < CHUNK; ++tt) {
    const int t = t0 + tt;
    if (tid < 64) kv[tid] = kb[(size_t)t * M_ + tid];
    else if (tid < 128) vv_[tid - 64] = vb[(size_t)t * M_ + tid - 64];
    else if (tid < 192) qv[tid - 128] = qb[(size_t)t * M_ + tid - 128];
    __syncthreads();

    // u = Inv * k
    if (tid < 64) {
      float s = 0.f;
#pragma unroll 8
      for (int j = 0; j < 64; ++j) s += Inv[tid][j] * kv[j];
      uu[tid] = s;
    }
    __syncthreads();
    // denom = 1 + k.u ; vdot = v.v
    if (tid < 64) red[tid] = kv[tid] * uu[tid];
    else if (tid < 128) red[tid] = vv_[tid - 64] * vv_[tid - 64];
    __syncthreads();
    for (int off = 32; off > 0; off >>= 1) {
      if (tid < off) red[tid] += red[tid + off];
      else if (tid >= 64 && tid < 64 + off) red[tid] += red[tid + off];
      __syncthreads();
    }
    const float rdenom = 1.f / (1.f + red[0]);
    if (tid == 0) sc[0] += red[64];     // running trace(C_vv)
    __syncthreads();

    // rank-1 state updates + Sherman-Morrison inverse update
    for (int e = tid; e < 64 * 64; e += 256) {
      int r = e >> 6, cc = e & 63;
      Ckk[r][cc] += kv[r] * kv[cc];
      Cvk[r][cc] += vv_[r] * kv[cc];
      Inv[r][cc] -= uu[r] * uu[cc] * rdenom;
    }
    __syncthreads();

    mm64(Cvk, Inv, Mem, tid);           // memory = C_vk * inv
    __syncthreads();
    mm64(Mem, Ckk, Tmp, tid);           // tmp = memory * C_kk
    __syncthreads();

    float pa = 0.f, pp = 0.f;
    for (int e = tid; e < 64 * 64; e += 256) {
      int r = e >> 6, cc = e & 63;
      float m = Mem[r][cc];
      pa += m * Cvk[r][cc];
      pp += m * Tmp[r][cc];
    }
    red[tid] = pa;
    __syncthreads();
    for (int off = 128; off > 0; off >>= 1) {
      if (tid < off) red[tid] += red[tid + off];
      __syncthreads();
    }
    const float align = red[0];
    __syncthreads();
    red[tid] = pp;
    __syncthreads();
    for (int off = 128; off > 0; off >>= 1) {
      if (tid < off) red[tid] += red[tid + off];
      __syncthreads();
    }
    const float pred = red[0];
    const float targ = sc[0];
    lsum += align / (sqrtf(pred * targ) + 1e-8f);

    // out_x[t] = memory * q
    if (tid < 64) {
      float s = 0.f;
#pragma unroll 8
      for (int j = 0; j < 64; ++j) s += Mem[tid][j] * qv[j];
      outx[((size_t)(b * T_ + t)) * M_ + tid] = s;
    }
    __syncthreads();
  }
  if (tid == 0) lpart[(size_t)b * NC + c] = lsum;
}

// ============================================================
// Kernel 6: final loss reduction (B*NC == 256 partials)
// ============================================================
__global__ __launch_bounds__(256) void loss_kernel(const float* __restrict__ lpart,
                                                   float* __restrict__ out) {
  __shared__ float red[256];
  const int tid = threadIdx.x;
  red[tid] = lpart[tid];
  __syncthreads();
  for (int off = 128; off > 0; off >>= 1) {
    if (tid < off) red[tid] += red[tid + off];
    __syncthreads();
  }
  if (tid == 0) out[LOSS_OFF] = -red[0] / (float)(B_ * T_);
}

// ============================================================
// Kernel 7: y = x + out_x @ out_w^T + out_b  via fp32 WMMA
// grid (T/16, D/64, B), block 128 (4 waves)
// ============================================================
__global__ __launch_bounds__(128) void yout_kernel(const float* __restrict__ x,
                                                   const float* __restrict__ outx,
                                                   const float* __restrict__ ow,
                                                   const float* __restrict__ ob,
                                                   float* __restrict__ y) {
  const int t0 = blockIdx.x * 16;
  const int n0 = blockIdx.y * 64;
  const int b = blockIdx.z;
  const int tid = threadIdx.x;
  const int wv = tid >> 5, lane = tid & 31;
  const int row = lane & 15, hi = lane >> 4;

  __shared__ float sa[16][68];   // out_x tile  [t][m]
  __shared__ float sb[64][68];   // sb[m][n] = ow[n0+n][m]
  for (int e = tid; e < 16 * 64; e += 128) {
    int r = e >> 6, m = e & 63;
    sa[r][m] = outx[((size_t)(b * T_ + t0 + r)) * M_ + m];
  }
  for (int e = tid; e < 64 * 64; e += 128) {
    int n = e >> 6, m = e & 63;
    sb[m][n] = ow[(size_t)(n0 + n) * M_ + m];
  }
  __syncthreads();

  v8f acc = {};
#pragma unroll
  for (int kq = 0; kq < 16; ++kq) {
    const int k0 = kq * 4 + 2 * hi;
    v2f a;
    a.x = sa[row][k0];
    a.y = sa[row][k0 + 1];
    v2f bb;
    bb.x = sb[k0][wv * 16 + row];
    bb.y = sb[k0 + 1][wv * 16 + row];
    acc = wmma4(a, bb, acc);
  }
#pragma unroll
  for (int r = 0; r < 8; ++r) {
    const int t = t0 + r + 8 * hi;
    const int d = n0 + wv * 16 + row;
    const size_t idx = ((size_t)(b * T_ + t)) * D_ + d;
    y[idx] = acc[r] + x[idx] + ob[d];
  }
}

extern "C" void kernel_launch(void* const* d_in, const int* in_sizes, int n_in,
                              void* d_out, int out_size, void* d_ws, size_t ws_size,
                              hipStream_t stream) {
  (void)in_sizes; (void)n_in; (void)out_size; (void)ws_size;
  const float* x = (const float*)d_in[0];
  const float* ln_gamma = (const float*)d_in[1];
  const float* ln_beta = (const float*)d_in[2];
  const float* qkv_lerp_w = (const float*)d_in[3];
  const float* qkv_w = (const float*)d_in[4];
  // d_in[5] (r_lerp_w) and d_in[6] (r_w) feed only `r`, which is dead code
  const float* out_w = (const float*)d_in[7];
  const float* out_b = (const float*)d_in[8];

  float* out = (float*)d_out;
  float* ws = (float*)d_ws;
  float* norm = ws + NORM_OFF;
  float* outx = ws + OUTX_OFF;
  float* sums = ws + SUMS_OFF;
  float* lpart = ws + LPART_OFF;

  ln_kernel<<<B_ * TP1, 256, 0, stream>>>(x, ln_gamma, ln_beta, norm);
  qkv_kernel<<<dim3(T_ / 16, B_, 3), 128, 0, stream>>>(norm, qkv_lerp_w, qkv_w, out);
  csum_kernel<<<dim3(NC, B_, 3), 128, 0, stream>>>(out, sums);
  scan_kernel<<<dim3(3, B_), 256, 0, stream>>>(sums, out);
  pert_kernel<<<dim3(NC, B_), 256, 0, stream>>>(out, sums, outx, lpart);
  loss_kernel<<<1, 256, 0, stream>>>(lpart, out);
  yout_kernel<<<dim3(T_ / 16, D_ / 64, B_), 128, 0, stream>>>(x, outx, out_w, out_b, out);
}